// modeler_Nc_19189913879149
// MI455X (gfx1250) — compile-verified
//
#include <hip/hip_runtime.h>
#include <hip/hip_bf16.h>

// Problem constants (match reference)
#define K_ASP 4
#define NN    50000
#define DD    64
#define EE    400000
#define LL    16
#define TAU_F 0.5f
#define BN_EPS 1e-5f

typedef __attribute__((ext_vector_type(2))) float v2f;
typedef __attribute__((ext_vector_type(8))) float v8f;

// Guaranteed-hardware fp32 atomic add (validated by the integrated assembler).
__device__ __forceinline__ void atomic_fadd_f32(float* p, float v) {
  asm volatile("global_atomic_add_f32 %0, %1, off" : : "v"(p), "v"(v) : "memory");
}

// ---------------------------------------------------------------- utilities
__global__ void zero_kernel(float* __restrict__ p, int n) {
  int i = blockIdx.x * blockDim.x + threadIdx.x;
  if (i < n) p[i] = 0.0f;
}

// ---------------------------------------------------------------- degrees
__global__ void degree_kernel(const int* __restrict__ src, const int* __restrict__ dst,
                              float* __restrict__ outdeg, float* __restrict__ indeg) {
  int e = blockIdx.x * blockDim.x + threadIdx.x;
  if (e < EE) {
    atomic_fadd_f32(&outdeg[src[e]], 1.0f);
    atomic_fadd_f32(&indeg[dst[e]], 1.0f);
  }
}

__global__ void norm_kernel(float* __restrict__ deg_src, float* __restrict__ deg_dst) {
  int n = blockIdx.x * blockDim.x + threadIdx.x;
  if (n < NN) {
    deg_src[n] = rsqrtf(fmaxf(deg_src[n], 1.0f));   // out_deg^-0.5 (clamped to >=1)
    deg_dst[n] = rsqrtf(fmaxf(deg_dst[n], 1.0f));   // in_deg^-0.5
  }
}

// ---------------------------------------------------------------- edge weights
// One wave (32 lanes) per edge; lane owns dims {2l, 2l+1}.
__global__ void edge_weight_kernel(const float* __restrict__ asp,     // [K*N, D]
                                   const float* __restrict__ attn_w,  // [1, D]
                                   const float* __restrict__ attn_b,  // [1]
                                   const float* __restrict__ gumbel,  // [E, K]
                                   const int*   __restrict__ w,       // [E, 2]
                                   float* __restrict__ ew_out) {      // [E, D]
  int gid  = blockIdx.x * blockDim.x + threadIdx.x;
  int e    = gid >> 5;
  int lane = threadIdx.x & 31;
  if (e >= EE) return;                              // wave-uniform exit

  int w0 = __builtin_amdgcn_readfirstlane(w[2 * e + 0]);
  int w1 = __builtin_amdgcn_readfirstlane(w[2 * e + 1]);
  int d0 = lane << 1;
  float awx = attn_w[d0], awy = attn_w[d0 + 1];

  float ewx[K_ASP], ewy[K_ASP], logit[K_ASP];
#pragma unroll
  for (int k = 0; k < K_ASP; ++k) {
    float2 a = *(const float2*)(asp + ((size_t)k * NN + (size_t)w0) * DD + d0);
    float2 b = *(const float2*)(asp + ((size_t)k * NN + (size_t)w1) * DD + d0);
    float ex = a.x + b.x, ey = a.y + b.y;
    ewx[k] = ex; ewy[k] = ey;
    float dv = ex * awx + ey * awy;
#pragma unroll
    for (int off = 16; off > 0; off >>= 1) dv += __shfl_xor(dv, off, 32);
    logit[k] = dv;                                   // replicated on all lanes
  }

  float ab = attn_b[0];
  float z[K_ASP], m = -3.4e38f;
#pragma unroll
  for (int k = 0; k < K_ASP; ++k) {
    z[k] = (logit[k] + ab + gumbel[e * K_ASP + k]) * (1.0f / TAU_F);
    m = fmaxf(m, z[k]);
  }
  float se = 0.0f;
#pragma unroll
  for (int k = 0; k < K_ASP; ++k) { z[k] = expf(z[k] - m); se += z[k]; }
  float inv = 1.0f / se;

  float ox = 0.0f, oy = 0.0f;
#pragma unroll
  for (int k = 0; k < K_ASP; ++k) {
    float a = z[k] * inv;
    ox += a * ewx[k];
    oy += a * ewy[k];
  }
  float2 o; o.x = ox; o.y = oy;
  *(float2*)(ew_out + (size_t)e * DD + d0) = o;
}

// ---------------------------------------------------------------- WMMA GEMM
// Exact fp32 on the matrix pipe: V_WMMA_F32_16X16X4_F32.
// A 16x4 f32 (ISA 7.12.2): lane (half,col) holds A[row=col][K=4kk+2*half+{0,1}]
//   -> a contiguous float2 of the A row.
// B 4x16 f32 (mirror of A): lane holds B[K=4kk+2*half+{0,1}][N=col].
// C/D 16x16 f32: VGPR i -> M = i + 8*half, N = col.
__global__ void gemm64_wmma_kernel(const float* __restrict__ A,
                                   const float* __restrict__ scale,
                                   const float* __restrict__ W,
                                   float* __restrict__ out) {
  int wave = (blockIdx.x * blockDim.x + threadIdx.x) >> 5;
  if (wave >= NN / 16) return;                       // wave-uniform exit
  int lane = threadIdx.x & 31;
  int half = lane >> 4;                              // 0: lanes 0-15, 1: lanes 16-31
  int col  = lane & 15;
  int row  = wave * 16 + col;
  float s  = scale[row];
  const float* arow = A + (size_t)row * DD;

  v8f acc[4] = {};
#pragma unroll
  for (int kk = 0; kk < 16; ++kk) {
    int kb = kk * 4 + half * 2;
    float2 av = *(const float2*)(arow + kb);
    v2f a; a[0] = av.x * s; a[1] = av.y * s;
#pragma unroll
    for (int nt = 0; nt < 4; ++nt) {
      v2f b;
      b[0] = W[(size_t)kb * DD + nt * 16 + col];
      b[1] = W[(size_t)(kb + 1) * DD + nt * 16 + col];
      acc[nt] = __builtin_amdgcn_wmma_f32_16x16x4_f32(
          false, a, false, b, (short)0, acc[nt], false, false);
    }
  }
#pragma unroll
  for (int nt = 0; nt < 4; ++nt)
#pragma unroll
    for (int i = 0; i < 8; ++i)
      out[(size_t)(wave * 16 + i + half * 8) * DD + nt * 16 + col] = acc[nt][i];
}

// out[N,16] = H[N,64] @ lin_w[16,64]^T + lin_b   (single 16-wide N tile)
__global__ void gemm_out_wmma_kernel(const float* __restrict__ H,
                                     const float* __restrict__ linw,
                                     const float* __restrict__ linb,
                                     float* __restrict__ out) {
  int wave = (blockIdx.x * blockDim.x + threadIdx.x) >> 5;
  if (wave >= NN / 16) return;
  int lane = threadIdx.x & 31;
  int half = lane >> 4;
  int col  = lane & 15;
  int row  = wave * 16 + col;
  const float* arow = H + (size_t)row * DD;
  const float* brow = linw + (size_t)col * DD;       // lin_w[c, k] == B[k, c]

  v8f acc = {};
#pragma unroll
  for (int kk = 0; kk < 16; ++kk) {
    int kb = kk * 4 + half * 2;
    float2 av = *(const float2*)(arow + kb);
    float2 bv = *(const float2*)(brow + kb);
    v2f a; a[0] = av.x; a[1] = av.y;
    v2f b; b[0] = bv.x; b[1] = bv.y;
    acc = __builtin_amdgcn_wmma_f32_16x16x4_f32(
        false, a, false, b, (short)0, acc, false, false);
  }
  float bias = linb[col];
#pragma unroll
  for (int i = 0; i < 8; ++i)
    out[(size_t)(wave * 16 + i + half * 8) * LL + col] = acc[i] + bias;
}

// ---------------------------------------------------------------- scatter
// One wave per edge; lane owns a float2 of the D=64 row. src/dst forced
// uniform (SGPR) so loads use saddr form; 2 hardware fp32 atomics per lane.
__global__ void message_kernel(const float* __restrict__ hW,
                               const float* __restrict__ ew,
                               const int* __restrict__ src,
                               const int* __restrict__ dst,
                               float* __restrict__ agg) {
  int wid = (blockIdx.x * blockDim.x + threadIdx.x) >> 5;
  if (wid >= EE) return;                             // wave-uniform exit
  int lane = threadIdx.x & 31;
  int s  = __builtin_amdgcn_readfirstlane(src[wid]);
  int dn = __builtin_amdgcn_readfirstlane(dst[wid]);
  int d0 = lane << 1;
  float2 hv = *(const float2*)(hW + (size_t)s * DD + d0);
  float2 ev = *(const float2*)(ew + (size_t)wid * DD + d0);
  float* p = agg + (size_t)dn * DD + d0;
  atomic_fadd_f32(p,     hv.x * ev.x);
  atomic_fadd_f32(p + 1, hv.y * ev.y);
}

// ---------------------------------------------------------------- batch norm
// y = agg*norm_dst + bias computed on the fly; accumulate per-column sum/sumsq.
__global__ void stats_kernel(const float* __restrict__ agg,
                             const float* __restrict__ norm_dst,
                             const float* __restrict__ bias,
                             float* __restrict__ colsum,
                             float* __restrict__ colsumsq) {
  int tid = threadIdx.x;                             // blockDim = 256 = 4 rows x 64 cols
  int d   = tid & 63;
  float s = 0.0f, ss = 0.0f;
  for (int row = blockIdx.x * 4 + (tid >> 6); row < NN; row += gridDim.x * 4) {
    float y = agg[(size_t)row * DD + d] * norm_dst[row] + bias[d];
    s += y; ss += y * y;
  }
  __shared__ float sh[512];
  sh[tid] = s; sh[256 + tid] = ss;
  __syncthreads();
  if (tid < 64) {
    s  = sh[tid] + sh[tid + 64] + sh[tid + 128] + sh[tid + 192];
    ss = sh[256 + tid] + sh[256 + tid + 64] + sh[256 + tid + 128] + sh[256 + tid + 192];
    atomic_fadd_f32(&colsum[tid], s);
    atomic_fadd_f32(&colsumsq[tid], ss);
  }
}

__global__ void finalize_stats_kernel(const float* __restrict__ colsum,
                                      const float* __restrict__ colsumsq,
                                      float* __restrict__ mean,
                                      float* __restrict__ rstd) {
  int d = threadIdx.x;
  if (d < DD) {
    float mu  = colsum[d] * (1.0f / NN);
    float var = colsumsq[d] * (1.0f / NN) - mu * mu;
    mean[d] = mu;
    rstd[d] = rsqrtf(fmaxf(var, 0.0f) + BN_EPS);
  }
}

__global__ void bn_apply_kernel(const float* __restrict__ agg,
                                const float* __restrict__ norm_dst,
                                const float* __restrict__ bias,
                                const float* __restrict__ mean,
                                const float* __restrict__ rstd,
                                const float* __restrict__ gamma,
                                const float* __restrict__ beta,
                                float* __restrict__ out, int do_elu) {
  int idx = blockIdx.x * blockDim.x + threadIdx.x;   // N*D = 3.2M
  if (idx < NN * DD) {
    int d = idx & 63, row = idx >> 6;
    float y = agg[idx] * norm_dst[row] + bias[d];
    float x = (y - mean[d]) * rstd[d] * gamma[d] + beta[d];
    if (do_elu) x = (x > 0.0f) ? x : (expf(x) - 1.0f);
    out[idx] = x;
  }
}

// ---------------------------------------------------------------- launch
extern "C" void kernel_launch(void* const* d_in, const int* in_sizes, int n_in,
                              void* d_out, int out_size, void* d_ws, size_t ws_size,
                              hipStream_t stream) {
  (void)in_sizes; (void)n_in; (void)out_size; (void)ws_size;

  const float* aspect_emb = (const float*)d_in[0];   // [K*N, D]
  const float* center_emb = (const float*)d_in[1];   // [N, D]
  const float* attn_w     = (const float*)d_in[2];   // [1, D]
  const float* attn_b     = (const float*)d_in[3];   // [1]
  const float* W1 = (const float*)d_in[4];
  const float* b1 = (const float*)d_in[5];
  const float* W2 = (const float*)d_in[6];
  const float* b2 = (const float*)d_in[7];
  const float* W3 = (const float*)d_in[8];
  const float* b3 = (const float*)d_in[9];
  const float* bn_gamma = (const float*)d_in[10];
  const float* bn_beta  = (const float*)d_in[11];
  const float* lin_w    = (const float*)d_in[12];    // [L, D]
  const float* lin_b    = (const float*)d_in[13];    // [L]
  const float* gumbel   = (const float*)d_in[14];    // [E, K]
  const int*   w        = (const int*)d_in[15];      // [E, 2]
  const int*   src      = (const int*)d_in[16];      // [E]
  const int*   dst      = (const int*)d_in[17];      // [E]

  float* out        = (float*)d_out;
  float* h_final    = out;                           // [N, D]
  float* logits_out = out + (size_t)NN * DD;         // [N, L]

  // workspace layout (floats): ~141 MB total
  float* ws = (float*)d_ws;
  size_t off = 0;
  float* ew       = ws + off; off += (size_t)EE * DD;   // 25.6M
  float* hbuf     = ws + off; off += (size_t)NN * DD;   //  3.2M
  float* hW       = ws + off; off += (size_t)NN * DD;   //  3.2M
  float* agg      = ws + off; off += (size_t)NN * DD;   //  3.2M (stats follow!)
  float* colsum   = ws + off; off += 64;
  float* colsumsq = ws + off; off += 64;
  float* meanb    = ws + off; off += 64;
  float* rstdb    = ws + off; off += 64;
  float* norm_src = ws + off; off += NN;                // contiguous pair
  float* norm_dst = ws + off; off += NN;

  // ---- degrees -> norms
  zero_kernel<<<(2 * NN + 255) / 256, 256, 0, stream>>>(norm_src, 2 * NN);
  degree_kernel<<<(EE + 255) / 256, 256, 0, stream>>>(src, dst, norm_src, norm_dst);
  norm_kernel<<<(NN + 255) / 256, 256, 0, stream>>>(norm_src, norm_dst);

  // ---- edge weights (one wave per edge)
  edge_weight_kernel<<<(EE * 32) / 256, 256, 0, stream>>>(
      aspect_emb, attn_w, attn_b, gumbel, w, ew);

  // ---- 3 GraphConv + BN (+ELU) layers
  const float* Ws[3] = {W1, W2, W3};
  const float* bs[3] = {b1, b2, b3};
  const float* hin = center_emb;
  const int gemm_blocks = ((NN / 16) * 32 + 127) / 128;
  for (int layer = 0; layer < 3; ++layer) {
    // zero agg + colsum + colsumsq (contiguous)
    zero_kernel<<<(NN * DD + 128 + 255) / 256, 256, 0, stream>>>(agg, NN * DD + 128);
    gemm64_wmma_kernel<<<gemm_blocks, 128, 0, stream>>>(hin, norm_src, Ws[layer], hW);
    message_kernel<<<(EE * 32) / 256, 256, 0, stream>>>(hW, ew, src, dst, agg);
    stats_kernel<<<256, 256, 0, stream>>>(agg, norm_dst, bs[layer], colsum, colsumsq);
    finalize_stats_kernel<<<1, 64, 0, stream>>>(colsum, colsumsq, meanb, rstdb);
    float* hout = (layer == 2) ? h_final : hbuf;
    bn_apply_kernel<<<(NN * DD) / 256, 256, 0, stream>>>(
        agg, norm_dst, bs[layer], meanb, rstdb, bn_gamma, bn_beta, hout,
        (layer < 2) ? 1 : 0);
    hin = hout;
  }

  // ---- final projection [N,64] @ [64,16] + bias
  gemm_out_wmma_kernel<<<gemm_blocks, 128, 0, stream>>>(h_final, lin_w, lin_b, logits_out);
}